// FusedRejectionSampler_22445499089617
// MI455X (gfx1250) — compile-verified
//
#include <hip/hip_runtime.h>
#include <hip/hip_bf16.h>
#include <stdint.h>

// ---------------------------------------------------------------------------
// Fused rejection sampler for speculative decoding (MI455X / gfx1250).
// Phase 1: per-token streaming argmax of max(target-draft,0) over VOCAB,
//          staged through LDS with gfx1250 async global->LDS copies
//          (ASYNCcnt / s_wait_asynccnt), double-buffered, NT-hinted,
//          immediate-offset address generation (INST_OFFSET applies to both
//          the LDS dest and the global source, matching our slot layout).
// Phase 2: tiny per-batch finalize (ragged scatter + leading-accept count).
// ---------------------------------------------------------------------------

#define VOCAB       50257
#define MAX_SPEC    8
#define PLACEHOLDER (-1)
#define THREADS     256
#define EPT         8                          // elems / thread / tile / matrix
#define TILE        (THREADS * EPT)            // 2048 floats per matrix per stage
#define NTILES      ((VOCAB + TILE - 1) / TILE)  // 25

// Async global->LDS copy with compile-time byte offset. Data is streamed once
// (617 MB >> 192 MB L2) -> non-temporal hint.
template <int OFF>
__device__ __forceinline__ void async_ld_b32_off(uint32_t lds_off, uint64_t gaddr) {
  asm volatile("global_load_async_to_lds_b32 %0, %1, off offset:%2 th:TH_LOAD_NT"
               :: "v"(lds_off), "v"(gaddr), "n"(OFF)
               : "memory");
}

// Emit EPT transfers per matrix (2*EPT = 16 wave-level async instructions).
template <int K>
struct IssueK {
  static __device__ __forceinline__ void run(uint32_t lT, uint64_t gT,
                                             uint32_t lD, uint64_t gD) {
    IssueK<K - 1>::run(lT, gT, lD, gD);
    async_ld_b32_off<(K - 1) * 4>(lT, gT);
    async_ld_b32_off<(K - 1) * 4>(lD, gD);
  }
};
template <>
struct IssueK<0> {
  static __device__ __forceinline__ void run(uint32_t, uint64_t, uint32_t, uint64_t) {}
};

// Inline asm (not builtin) so the "memory" clobber is also a compiler barrier
// for the subsequent LDS reads.
#define WAIT_ASYNC_LE_TILE() asm volatile("s_wait_asynccnt 0x10" ::: "memory")
#define WAIT_ASYNC_ZERO()    asm volatile("s_wait_asynccnt 0x0"  ::: "memory")

// ---------------------------------------------------------------------------
// Kernel 1: one block per draft token. Streams the token's target/draft prob
// rows once, computes residual argmax + acceptance decision.
// ---------------------------------------------------------------------------
__global__ __launch_bounds__(THREADS)
void rs_token_kernel(const float* __restrict__ target_probs,
                     const float* __restrict__ draft_probs,
                     const int*   __restrict__ draft_token_ids,
                     const float* __restrict__ uniform_samples,
                     int* __restrict__ accepted_ws,
                     int* __restrict__ argmax_ws) {
  extern __shared__ float dyn[];       // 4 * TILE floats = 32 KB
  __shared__ float rv[THREADS];
  __shared__ int   ri[THREADS];

  const int tid = threadIdx.x;
  const int tok = blockIdx.x;

  const uint64_t trow =
      (uint64_t)(uintptr_t)target_probs + (uint64_t)tok * (uint64_t)VOCAB * 4ull;
  const uint64_t drow =
      (uint64_t)(uintptr_t)draft_probs  + (uint64_t)tok * (uint64_t)VOCAB * 4ull;

  // Workgroup-relative LDS byte offset of the dynamic-LDS staging region:
  // dynamic LDS starts immediately after the static allocations.
  const uint32_t dyn_off = __builtin_amdgcn_groupstaticsize();
  const uint32_t slot    = (uint32_t)(tid * EPT * 4);
  // per-thread staging slots: T buf0, T buf1, D buf0, D buf1
  const uint32_t lT0 = dyn_off + 0u * (TILE * 4u) + slot;
  const uint32_t lT1 = dyn_off + 1u * (TILE * 4u) + slot;
  const uint32_t lD0 = dyn_off + 2u * (TILE * 4u) + slot;
  const uint32_t lD1 = dyn_off + 3u * (TILE * 4u) + slot;

  // Clamped per-thread element base for tile t: all EPT elements in-row.
  // Tail threads get overlapping windows (duplicates are harmless for argmax).
  auto clamped_base = [&](int t) {
    int b = t * TILE + tid * EPT;
    return (b > VOCAB - EPT) ? (VOCAB - EPT) : b;
  };

  // Issue one tile (16 wave-level async instructions, full EXEC, so per-wave
  // ASYNCcnt increments are deterministic for the s_wait thresholds).
  auto issue = [&](int t, int b) {
    const uint64_t boff = (uint64_t)clamped_base(t) * 4ull;
    IssueK<EPT>::run(b ? lT1 : lT0, trow + boff,
                     b ? lD1 : lD0, drow + boff);
  };

  float bv = -1.0f;   // residuals are >= 0, so any real element wins
  int   bi = 0;

  issue(0, 0);
#pragma unroll 1
  for (int t = 0; t < NTILES; ++t) {
    const int cur = t & 1;
    if (t + 1 < NTILES) {
      issue(t + 1, cur ^ 1);
      WAIT_ASYNC_LE_TILE();   // <=16 outstanding => tile t fully landed in LDS
    } else {
      WAIT_ASYNC_ZERO();
    }
    // generic-pointer views of the staging buffers (runtime selects only —
    // no constant aggregates referencing the LDS symbol)
    float* bT = cur ? (dyn + TILE)     : dyn;
    float* bD = cur ? (dyn + 3 * TILE) : (dyn + 2 * TILE);
    const int cb = clamped_base(t);
#pragma unroll
    for (int k = 0; k < EPT; ++k) {
      float r = bT[tid * EPT + k] - bD[tid * EPT + k];
      r = (r > 0.0f) ? r : 0.0f;
      if (r > bv) { bv = r; bi = cb + k; }   // ascending scan -> first-max kept
    }
  }

  // Block-wide argmax reduction (value desc, index asc tie-break == jnp.argmax).
  rv[tid] = bv;
  ri[tid] = bi;
  __syncthreads();
#pragma unroll
  for (int s = THREADS / 2; s > 0; s >>= 1) {
    if (tid < s) {
      const float ov = rv[tid + s];
      const int   oi = ri[tid + s];
      if (ov > rv[tid] || (ov == rv[tid] && oi < ri[tid])) {
        rv[tid] = ov;
        ri[tid] = oi;
      }
    }
    __syncthreads();
  }

  if (tid == 0) {
    const int id  = draft_token_ids[tok];
    const float pd = draft_probs[(size_t)tok * VOCAB + id];
    const float pt = target_probs[(size_t)tok * VOCAB + id];
    const float acc = fminf(pt / fmaxf(pd, 1e-10f), 1.0f);
    accepted_ws[tok] = (uniform_samples[tok] < acc) ? 1 : 0;
    argmax_ws[tok]   = ri[0];
  }
}

// ---------------------------------------------------------------------------
// Kernel 2: per-batch-row finalize. One block of 256 threads (batch == 256).
// ---------------------------------------------------------------------------
__global__ __launch_bounds__(THREADS)
void rs_finalize_kernel(const int* __restrict__ draft_token_ids,
                        const int* __restrict__ num_draft_tokens,
                        const int* __restrict__ bonus_token_ids,
                        const int* __restrict__ accepted_ws,
                        const int* __restrict__ argmax_ws,
                        int* __restrict__ out, int batch) {
  __shared__ int lens[THREADS];
  const int b = threadIdx.x;
  lens[b] = (b < batch) ? num_draft_tokens[b] : 0;
  __syncthreads();
  if (b >= batch) return;

  int start = 0;
  for (int i = 0; i < b; ++i) start += lens[i];   // cu[b]
  const int len = lens[b];

  int cnt = 0;                                    // leading accepts
  while (cnt < len && accepted_ws[start + cnt] != 0) ++cnt;

  const bool all_acc  = (cnt >= len);
  // not-all-accepted => cnt <= len-1 <= MAX_SPEC-1, so first_reject == cnt
  const int final_tok = all_acc ? bonus_token_ids[b] : argmax_ws[start + cnt];

  int* output          = out;                              // [batch, MAX_SPEC+1]
  int* num_accepted    = out + batch * (MAX_SPEC + 1);
  int* accepted_counts = num_accepted + batch;
  int* recovered_cnts  = accepted_counts + batch;
  int* bonus_cnts      = recovered_cnts + batch;

#pragma unroll
  for (int j = 0; j < MAX_SPEC + 1; ++j) {
    int v = PLACEHOLDER;
    if (j < cnt)       v = draft_token_ids[start + j];
    else if (j == cnt) v = final_tok;
    output[b * (MAX_SPEC + 1) + j] = v;
  }
  num_accepted[b]    = cnt + 1;
  accepted_counts[b] = cnt;
  recovered_cnts[b]  = all_acc ? 0 : 1;
  bonus_cnts[b]      = all_acc ? 1 : 0;
}

// ---------------------------------------------------------------------------
// Host entry
// ---------------------------------------------------------------------------
extern "C" void kernel_launch(void* const* d_in, const int* in_sizes, int n_in,
                              void* d_out, int out_size, void* d_ws, size_t ws_size,
                              hipStream_t stream) {
  const int*   draft_token_ids  = (const int*)d_in[0];
  const int*   num_draft_tokens = (const int*)d_in[1];
  const float* draft_probs      = (const float*)d_in[2];
  const float* target_probs     = (const float*)d_in[3];
  const int*   bonus_token_ids  = (const int*)d_in[4];
  const float* uniform_samples  = (const float*)d_in[5];

  const int num_tokens = in_sizes[0];
  const int batch      = in_sizes[1];

  int* accepted_ws = (int*)d_ws;
  int* argmax_ws   = accepted_ws + num_tokens;

  const size_t lds_bytes = (size_t)(4 * TILE) * sizeof(float);  // 32 KB
  rs_token_kernel<<<num_tokens, THREADS, lds_bytes, stream>>>(
      target_probs, draft_probs, draft_token_ids, uniform_samples,
      accepted_ws, argmax_ws);

  rs_finalize_kernel<<<1, THREADS, 0, stream>>>(
      draft_token_ids, num_draft_tokens, bonus_token_ids,
      accepted_ws, argmax_ws, (int*)d_out, batch);
}